// ROIPooling_90357521973497
// MI455X (gfx1250) — compile-verified
//
#include <hip/hip_runtime.h>
#include <hip/hip_bf16.h>

// ROI pooling = pure gather of 2048-byte channel vectors.
//   Kernel 1: per-proposal nearest-neighbor source indices (rows[1000][7], cols[1000][7]).
//   Kernel 2: one wave32 per output cell copies fm[row,col,0:512] -> out, 16B/lane B128.
// CDNA5 path: async global->LDS->global (ASYNCcnt) — builtins confirmed present;
// they take int4* (vector_size(16)) pointers per the round-1 diagnostic.

#define POOLSZ   7
#define NPROP    1000
#define FM_H     128
#define FM_W     128
#define FM_C     512
#define NVEC     (NPROP * POOLSZ * POOLSZ)   // 49000 channel vectors
#define WPB      8                            // waves per block
#define TPB      (WPB * 32)

#define HAS_ASYNC_LDS ( __has_builtin(__builtin_amdgcn_global_load_async_to_lds_b128) && \
                        __has_builtin(__builtin_amdgcn_global_store_async_from_lds_b128) && \
                        __has_builtin(__builtin_amdgcn_s_wait_asynccnt) )

typedef float f4  __attribute__((ext_vector_type(4)));
typedef int   v4i __attribute__((vector_size(16)));          // param type the builtins want
typedef __attribute__((address_space(1))) v4i* gv4i_ptr;     // global int4*
typedef __attribute__((address_space(3))) v4i* lv4i_ptr;     // LDS int4*

// ---------------------------------------------------------------------------
// Kernel 1: index computation (matches jnp reference exactly in f32 arithmetic)
// ---------------------------------------------------------------------------
__global__ __launch_bounds__(256) void roi_idx_kernel(
    const float* __restrict__ props,   // [NPROP][4] : xmin,ymin,xmax,ymax (float)
    int* __restrict__ rows,            // [NPROP][7]
    int* __restrict__ cols)            // [NPROP][7]
{
  int p = blockIdx.x * blockDim.x + threadIdx.x;
  if (p >= NPROP) return;

  float x0 = props[p * 4 + 0];
  float y0 = props[p * 4 + 1];
  float x1 = props[p * 4 + 2];
  float y1 = props[p * 4 + 3];

  int xmin = (int)floorf(x0);
  int ymin = (int)floorf(y0);
  int xmax = (int)ceilf(x1);
  int ymax = (int)ceilf(y1);

  int spanY = ymax - ymin;
  int spanX = xmax - xmin;

#pragma unroll
  for (int i = 0; i < POOLSZ; ++i) {
    float c = (i + 0.5f) / 7.0f;           // half-pixel centers, same expr as jnp

    int offY = (int)floorf(c * (float)spanY);
    offY = min(offY, spanY - 1);
    int ry  = ymin + offY;
    ry      = max(0, min(ry, FM_H - 1));
    rows[p * POOLSZ + i] = ry;

    int offX = (int)floorf(c * (float)spanX);
    offX = min(offX, spanX - 1);
    int cx  = xmin + offX;
    cx      = max(0, min(cx, FM_W - 1));
    cols[p * POOLSZ + i] = cx;
  }
}

// ---------------------------------------------------------------------------
// Kernel 2: gather/copy. One wave per (proposal, ph, pw): 2048B, 4 x B128/lane.
// ---------------------------------------------------------------------------
__global__ __launch_bounds__(TPB) void roi_gather_kernel(
    const float* __restrict__ fm,      // [FM_H][FM_W][FM_C]
    const int* __restrict__ rows,      // [NPROP][7]
    const int* __restrict__ cols,      // [NPROP][7]
    float* __restrict__ out)           // [NVEC][FM_C]
{
  const int wave = threadIdx.x >> 5;
  const int lane = threadIdx.x & 31;

  int v = blockIdx.x * WPB + wave;     // vector id: p*49 + ph*7 + pw (wave-uniform)
  if (v >= NVEC) return;               // whole wave exits together (v uniform)

  int p  = v / 49;
  int r  = v - p * 49;
  int ph = r / 7;
  int pw = r - ph * 7;

  int row = rows[p * POOLSZ + ph];
  int col = cols[p * POOLSZ + pw];

  const float* src = fm  + ((size_t)(row * FM_W + col)) * FM_C;  // 2048B-aligned
  float*       dst = out + (size_t)v * FM_C;                     // 2048B-aligned

#if HAS_ASYNC_LDS
  // CDNA5 async data-mover path: HBM -> LDS -> HBM, no VGPR staging.
  // INST_OFFSET applies to BOTH global and LDS addresses (ISA 15.18 pseudocode),
  // so four issues at offsets {0,512,1024,1536} move the full 2KB vector.
  __shared__ float stage[WPB * FM_C];                 // 16 KB / block
  float* myLds = &stage[wave * FM_C];

  gv4i_ptr gsrc = (gv4i_ptr)(src   + lane * 4);
  gv4i_ptr gdst = (gv4i_ptr)(dst   + lane * 4);
  lv4i_ptr lbuf = (lv4i_ptr)(myLds + lane * 4);

  __builtin_amdgcn_global_load_async_to_lds_b128(gsrc, lbuf,    0, 0);
  __builtin_amdgcn_global_load_async_to_lds_b128(gsrc, lbuf,  512, 0);
  __builtin_amdgcn_global_load_async_to_lds_b128(gsrc, lbuf, 1024, 0);
  __builtin_amdgcn_global_load_async_to_lds_b128(gsrc, lbuf, 1536, 0);
  __builtin_amdgcn_s_wait_asynccnt(0);               // loads landed in LDS

  __builtin_amdgcn_global_store_async_from_lds_b128(gdst, lbuf,    0, 0);
  __builtin_amdgcn_global_store_async_from_lds_b128(gdst, lbuf,  512, 0);
  __builtin_amdgcn_global_store_async_from_lds_b128(gdst, lbuf, 1024, 0);
  __builtin_amdgcn_global_store_async_from_lds_b128(gdst, lbuf, 1536, 0);
  __builtin_amdgcn_s_wait_asynccnt(0);               // LDS safe to reuse
#else
  // Fallback: B128 load -> non-temporal B128 store (output is streamed once;
  // keep the 32MB feature map resident in the 192MB L2 with default-RT loads).
  const f4* s4 = (const f4*)src;
  f4*       d4 = (f4*)dst;
#pragma unroll
  for (int k = 0; k < 4; ++k) {
    f4 t = s4[lane + 32 * k];
    __builtin_nontemporal_store(t, &d4[lane + 32 * k]);
  }
#endif
}

// ---------------------------------------------------------------------------
// Host launcher
// ---------------------------------------------------------------------------
extern "C" void kernel_launch(void* const* d_in, const int* in_sizes, int n_in,
                              void* d_out, int out_size, void* d_ws, size_t ws_size,
                              hipStream_t stream) {
  const float* fm    = (const float*)d_in[0];   // (1,128,128,512) f32
  const float* props = (const float*)d_in[1];   // (1,1000,4)      f32
  float* out = (float*)d_out;                   // (1,1000,7,7,512) f32

  int* rows = (int*)d_ws;                       // [1000][7]
  int* cols = rows + NPROP * POOLSZ;            // [1000][7]

  roi_idx_kernel<<<(NPROP + 255) / 256, 256, 0, stream>>>(props, rows, cols);

  roi_gather_kernel<<<NVEC / WPB, TPB, 0, stream>>>(fm, rows, cols, out);
}